// ArbitraryBatchTimeSeriesInterpolator_49563922596772
// MI455X (gfx1250) — compile-verified
//
#include <hip/hip_runtime.h>
#include <stdint.h>

// Problem constants (from reference)
#define NTIME  4096
#define NBATCH 4096
#define NQ     8192

// Tiling
#define CS      32               // coarse stride (knots per coarse cell)
#define NC      (NTIME / CS)     // 128 coarse knots per column
#define BB      64               // batch columns per block
#define QPB     128              // queries per block
#define THREADS 256

__global__ __launch_bounds__(THREADS)
void interp_kernel(const float* __restrict__ times,
                   const float* __restrict__ values,
                   const float* __restrict__ tq,
                   float* __restrict__ out)
{
    // Coarse time-knot table: sCoarse[k*BB + c] = times[k*CS, b0+c]
    __shared__ float sCoarse[NC * BB];   // 32 KB

    const int tid = threadIdx.x;
    const int b0  = blockIdx.x * BB;
    const int qb  = blockIdx.y * QPB;

    // ---- Stage coarse knots into LDS via CDNA5 async loads (ASYNCcnt path) ----
    // element e -> (k = e/64, c = e%64); 32 elements per thread.
#pragma unroll
    for (int j = 0; j < (NC * BB) / THREADS; ++j) {
        int e = tid + j * THREADS;
        int k = e >> 6;
        int c = e & (BB - 1);
        unsigned int lds_off =
            (unsigned int)(uintptr_t)(&sCoarse[e]);          // low 32 bits = LDS offset
        unsigned long long gaddr =
            (unsigned long long)(uintptr_t)(times + (size_t)k * CS * NBATCH + b0 + c);
        asm volatile("global_load_async_to_lds_b32 %0, %1, off"
                     :: "v"(lds_off), "v"(gaddr) : "memory");
    }
    asm volatile("s_wait_asynccnt 0x0" ::: "memory");
    __syncthreads();

    const int c = tid & (BB - 1);   // column within tile; wave covers 32 consecutive c
    const int r = tid >> 6;         // 0..3 query-row group
    const int b = b0 + c;

#pragma unroll 4
    for (int k = 0; k < QPB / 4; ++k) {        // 32 queries per thread
        const int q = qb + r * (QPB / 4) + k;  // lanes of a wave share q -> coalesced
        const float tv = tq[(size_t)q * NBATCH + b];

        // ---- Coarse lower bound in LDS: cc = #{ coarse[k] < tv }, 0..128 ----
        int cc = 0;
#pragma unroll
        for (int step = 128; step >= 1; step >>= 1) {
            int np = cc + step;
            if (np <= NC) {
                float cm = sCoarse[(np - 1) * BB + c];  // bank = c, conflict-free
                cc = (cm < tv) ? np : cc;
            }
        }

        // ---- Fine lower bound in global (L2-resident), uniform 32-wide window ----
        // i = first index with times[i] >= tv (i == NTIME if none)
        int i;
        if (cc == 0) {
            i = 0;                                // tv <= times[0]
        } else {
            const int L = (cc - 1) * CS + 1;      // lower bound lies in [L, L+31]
            int off = 0;
#pragma unroll
            for (int step = 16; step >= 1; step >>= 1) {
                int np = off + step;
                if (np <= 31) {
                    float tm = times[(size_t)(L + np - 1) * NBATCH + b];
                    off = (tm < tv) ? np : off;
                }
            }
            i = L + off;
        }

        int idx = i - 1;
        idx = idx < 0 ? 0 : idx;
        idx = idx > (NTIME - 2) ? (NTIME - 2) : idx;

        const size_t o0 = (size_t)idx * NBATCH + b;
        const float t0 = times[o0];
        const float t1 = times[o0 + NBATCH];
        const float v0 = values[o0];
        const float v1 = values[o0 + NBATCH];

        const float s = (v1 - v0) * __builtin_amdgcn_rcpf(t1 - t0);
        out[(size_t)q * NBATCH + b] = fmaf(s, tv - t0, v0);
    }
}

extern "C" void kernel_launch(void* const* d_in, const int* in_sizes, int n_in,
                              void* d_out, int out_size, void* d_ws, size_t ws_size,
                              hipStream_t stream) {
    const float* times  = (const float*)d_in[0];
    const float* values = (const float*)d_in[1];
    const float* t      = (const float*)d_in[2];
    float* out          = (float*)d_out;

    dim3 grid(NBATCH / BB, NQ / QPB);   // 64 x 64 blocks
    interp_kernel<<<grid, dim3(THREADS), 0, stream>>>(times, values, t, out);
}